// DualPathFusion_6365141533186
// MI455X (gfx1250) — compile-verified
//
#include <hip/hip_runtime.h>

typedef __attribute__((ext_vector_type(2))) float v2f;
typedef __attribute__((ext_vector_type(8))) float v8f;

#define Bc 2
#define Cc 32
#define Sc (32 * 128 * 128)          /* 524288 spatial sites per batch      */
#define TOTAL_SITES (Bc * Sc)        /* 1048576                             */
#define TILES (TOTAL_SITES / 16)     /* 65536 16-site WMMA tiles            */
#define TPW 8                        /* tiles per wave                      */
#define WAVES (TILES / TPW)          /* 8192                                */
#define BLOCK 256
#define GRID (WAVES / (BLOCK / 32))  /* 1024                                */

// logits(2 x S) = Wmat(2 x 64) @ [f1;f2](64 x S) via V_WMMA_F32_16X16X4_F32,
// then softmax over the 2 logits and per-channel blend from live registers.
__global__ __launch_bounds__(BLOCK) void dual_path_fusion_wmma(
    const float* __restrict__ f1, const float* __restrict__ f2,
    const float* __restrict__ w1, const float* __restrict__ b1v,
    const float* __restrict__ w2, const float* __restrict__ b2v,
    float* __restrict__ out) {
  // Folded weight matrix: rows 0/1 = a1/a2 weight rows over K=64, rows 2..15 = 0.
  __shared__ float wmat[16 * 64];
  for (int i = threadIdx.x; i < 16 * 64; i += BLOCK) {
    int m = i >> 6, k = i & 63;
    float v = 0.0f;
    if (m == 0) v = (k < 32) ? (w1[k] + 0.5f * w1[k + 32]) : (0.5f * w1[k]);
    if (m == 1) v = (k < 32) ? (0.5f * w2[k + 32]) : (w2[k - 32] + 0.5f * w2[k]);
    wmat[i] = v;
  }
  __syncthreads();

  const int lane = threadIdx.x & 31;
  const int m    = lane & 15;        // site-in-tile (B/C/D cols) / A row (M)
  const int hi   = lane >> 4;        // upper lane half -> K offset +2
  const int wave = blockIdx.x * (BLOCK / 32) + (threadIdx.x >> 5);

  // A operand per K-chunk t: VGPR0 = Wmat[m][4t + 2*hi], VGPR1 = Wmat[m][4t+1+2*hi]
  v2f A[16];
#pragma unroll
  for (int t = 0; t < 16; ++t) {
    A[t].x = wmat[m * 64 + 4 * t + 2 * hi];
    A[t].y = wmat[m * 64 + 4 * t + 1 + 2 * hi];
  }

  const float bias1 = b1v[0], bias2 = b2v[0];

  // Per-lane element offset inside a (C x S) slab: +2 channel rows for upper half.
  const unsigned laneOff = (unsigned)(2 * hi) * (unsigned)Sc + (unsigned)m;

  const size_t baseSite = (size_t)wave * (TPW * 16);
  const int b = (int)(baseSite / Sc);            // uniform per wave (128 | Sc)
  const size_t sIn   = baseSite - (size_t)b * Sc;
  const size_t slab  = (size_t)b * Cc * Sc;
  const float* base1 = f1 + slab + sIn;
  const float* base2 = f2 + slab + sIn;
  float*       baseo = out + slab + sIn;

  for (int it = 0; it < TPW; ++it) {
    const size_t s0 = (size_t)it * 16;

    // B operand: 64 x 16 tile. Chunks t=0..7 <- f1 channels, t=8..15 <- f2.
    v2f Bv[16];
#pragma unroll
    for (int t = 0; t < 8; ++t) {
      Bv[t].x     = base1[(size_t)(4 * t) * Sc + s0 + laneOff];
      Bv[t].y     = base1[(size_t)(4 * t + 1) * Sc + s0 + laneOff];
      Bv[t + 8].x = base2[(size_t)(4 * t) * Sc + s0 + laneOff];
      Bv[t + 8].y = base2[(size_t)(4 * t + 1) * Sc + s0 + laneOff];
    }

    // K=64 reduction: 16 chained fp32 WMMAs (D = A*B + C).
    v8f acc = {};
#pragma unroll
    for (int t = 0; t < 16; ++t)
      acc = __builtin_amdgcn_wmma_f32_16x16x4_f32(
          false, A[t], false, Bv[t], (short)0, acc, false, false);

    // D row0 -> a1 (acc[0], lanes 0-15), row1 -> a2 (acc[1], lanes 0-15).
    const float a1  = __shfl(acc[0], m, 32);
    const float a2  = __shfl(acc[1], m, 32);
    // softmax over {a1,a2}: wg1 = 1/(1+exp(a2-a1)); single v_rcp_f32 instead of
    // the IEEE v_div_scale/newton chain (exp path is already ~1 ulp).
    const float e   = __expf((a2 + bias2) - (a1 + bias1));
    const float wg1 = __builtin_amdgcn_rcpf(1.0f + e);
    const float wg2 = 1.0f - wg1;

    // Blend from live B registers: lane holding f1[c,s] (K=c) also holds
    // f2[c,s] (K=c+32). Store addresses = f1 load addresses into `out`.
#pragma unroll
    for (int t = 0; t < 8; ++t) {
      baseo[(size_t)(4 * t) * Sc + s0 + laneOff]     = Bv[t].x * wg1 + Bv[t + 8].x * wg2;
      baseo[(size_t)(4 * t + 1) * Sc + s0 + laneOff] = Bv[t].y * wg1 + Bv[t + 8].y * wg2;
    }
  }
}

extern "C" void kernel_launch(void* const* d_in, const int* in_sizes, int n_in,
                              void* d_out, int out_size, void* d_ws, size_t ws_size,
                              hipStream_t stream) {
  (void)in_sizes; (void)n_in; (void)out_size; (void)d_ws; (void)ws_size;
  const float* f1 = (const float*)d_in[0];
  const float* f2 = (const float*)d_in[1];
  const float* w1 = (const float*)d_in[2];
  const float* b1 = (const float*)d_in[3];
  const float* w2 = (const float*)d_in[4];
  const float* b2 = (const float*)d_in[5];
  float* outp = (float*)d_out;
  dual_path_fusion_wmma<<<GRID, BLOCK, 0, stream>>>(f1, f2, w1, b1, w2, b2, outp);
}